// CollisionAccuracy_15427522527884
// MI455X (gfx1250) — compile-verified
//
#include <hip/hip_runtime.h>
#include <hip/hip_bf16.h>
#include <math.h>

#define MAX_DIST_F 5.0f
#define EPS_F 1e-7f
#define CHUNK_BLKS 96   // 96 blocks * 32 lanes * 16B = 48KB LDS per chunk

typedef __attribute__((ext_vector_type(2))) float v2f;
typedef __attribute__((ext_vector_type(8))) float v8f;

// ---------------------------------------------------------------------------
// Kernel 0: zero workspace (normal accumulators + counts) and output.
// ---------------------------------------------------------------------------
__global__ void ca_init_kernel(float* acc, float* cnt, float* out,
                               int nAcc, int nCnt, int nOut) {
  int total = nAcc + nCnt + nOut;
  for (int i = blockIdx.x * blockDim.x + threadIdx.x; i < total;
       i += gridDim.x * blockDim.x) {
    if (i < nAcc)             acc[i] = 0.0f;
    else if (i < nAcc + nCnt) cnt[i - nAcc] = 0.0f;
    else                      out[i - nAcc - nCnt] = 0.0f;
  }
}

// ---------------------------------------------------------------------------
// Kernel 1: face normals, scatter-add into per-vertex accumulators + counts.
// ---------------------------------------------------------------------------
__global__ void ca_face_normals_kernel(const float* __restrict__ anchor,
                                       const int* __restrict__ faces,
                                       float* __restrict__ acc,
                                       float* __restrict__ cnt,
                                       int B, int Na, int F) {
  int idx = blockIdx.x * blockDim.x + threadIdx.x;
  if (idx >= B * F) return;
  int b = idx / F;
  int f = idx - b * F;

  int i0 = faces[3 * f + 0];
  int i1 = faces[3 * f + 1];
  int i2 = faces[3 * f + 2];

  const float* mb = anchor + (size_t)b * Na * 3;
  float v0x = mb[3 * i0 + 0], v0y = mb[3 * i0 + 1], v0z = mb[3 * i0 + 2];
  float e1x = mb[3 * i1 + 0] - v0x;
  float e1y = mb[3 * i1 + 1] - v0y;
  float e1z = mb[3 * i1 + 2] - v0z;
  float e2x = mb[3 * i2 + 0] - v0x;
  float e2y = mb[3 * i2 + 1] - v0y;
  float e2z = mb[3 * i2 + 2] - v0z;

  float fx = e1y * e2z - e1z * e2y;
  float fy = e1z * e2x - e1x * e2z;
  float fz = e1x * e2y - e1y * e2x;

  float* ab = acc + (size_t)b * Na * 3;
  atomicAdd(&ab[3 * i0 + 0], fx);
  atomicAdd(&ab[3 * i0 + 1], fy);
  atomicAdd(&ab[3 * i0 + 2], fz);
  atomicAdd(&ab[3 * i1 + 0], fx);
  atomicAdd(&ab[3 * i1 + 1], fy);
  atomicAdd(&ab[3 * i1 + 2], fz);
  atomicAdd(&ab[3 * i2 + 0], fx);
  atomicAdd(&ab[3 * i2 + 1], fy);
  atomicAdd(&ab[3 * i2 + 2], fz);

  if (b == 0) {
    atomicAdd(&cnt[i0], 1.0f);
    atomicAdd(&cnt[i1], 1.0f);
    atomicAdd(&cnt[i2], 1.0f);
  }
}

// ---------------------------------------------------------------------------
// Kernel 2: normalize vertex normals in place (acc becomes unit normals).
// ---------------------------------------------------------------------------
__global__ void ca_normalize_kernel(float* __restrict__ acc,
                                    const float* __restrict__ cnt,
                                    int B, int Na) {
  int idx = blockIdx.x * blockDim.x + threadIdx.x;
  if (idx >= B * Na) return;
  int b = idx / Na;
  int v = idx - b * Na;
  float c = cnt[v] + EPS_F;
  float* p = acc + ((size_t)b * Na + v) * 3;
  float nx = p[0] / c, ny = p[1] / c, nz = p[2] / c;
  float inv = 1.0f / (sqrtf(nx * nx + ny * ny + nz * nz) + EPS_F);
  p[0] = nx * inv;
  p[1] = ny * inv;
  p[2] = nz * inv;
}

// ---------------------------------------------------------------------------
// Kernel 2.5: pack anchors into WMMA-ready per-lane records.
// Record for (b, blk, lane): { B-matrix vgpr0, B-matrix vgpr1, a2, 0 }
//   col = lane&15, half = lane>>4; lower half K={x,y}, upper half K={z,0}.
// OOB anchors: B=0, a2=+huge so they never win the argmin.
// ---------------------------------------------------------------------------
__global__ void ca_pack_kernel(const float* __restrict__ anchor,
                               float4* __restrict__ pk,
                               int B, int Na, int nBlk) {
  int idx = blockIdx.x * blockDim.x + threadIdx.x;
  if (idx >= B * nBlk * 32) return;
  int lane = idx & 31;
  int t = idx >> 5;                 // b * nBlk + blk
  int b = t / nBlk;
  int blk = t - b * nBlk;
  int col = lane & 15;
  int half = lane >> 4;

  int aIdx = blk * 16 + col;
  bool valid = aIdx < Na;
  int aC = valid ? aIdx : (Na - 1);
  const float* ap = anchor + ((size_t)b * Na + aC) * 3;
  float ax = ap[0], ay = ap[1], az = ap[2];
  float a2 = valid ? (ax * ax + ay * ay + az * az) : 3.0e38f;
  float b0 = (half == 0) ? ax : az;
  float b1 = (half == 0) ? ay : 0.0f;
  if (!valid) { b0 = 0.0f; b1 = 0.0f; }
  pk[idx] = make_float4(b0, b1, a2, 0.0f);
}

// ---------------------------------------------------------------------------
// Cooperative global->LDS staging. On gfx1250 use the async-DMA path via
// inline asm (GLOBAL_LOAD_ASYNC_TO_LDS_B128, tracked by ASYNCcnt): the VDST
// operand is the LDS byte offset in a VGPR (low 32 bits of a generic pointer
// to a __shared__ object ARE the LDS offset; the aperture is in addr[63:32]),
// and the source is a 64-bit global address VGPR pair.
// ---------------------------------------------------------------------------
__device__ __forceinline__ void ca_stage_chunk(const float4* gsrc,
                                               float4* smem,
                                               int nEnt) {
#if defined(__gfx1250__)
  for (int e = threadIdx.x; e < nEnt; e += blockDim.x) {
    unsigned int loff = (unsigned int)(uintptr_t)(smem + e);   // LDS offset
    unsigned long long ga = (unsigned long long)(uintptr_t)(gsrc + e);
    asm volatile("global_load_async_to_lds_b128 %0, %1, off"
                 :: "v"(loff), "v"(ga)
                 : "memory");
  }
#if __has_builtin(__builtin_amdgcn_s_wait_asynccnt)
  __builtin_amdgcn_s_wait_asynccnt(0);
#else
  asm volatile("s_wait_asynccnt 0x0" ::: "memory");
#endif
#else
  for (int e = threadIdx.x; e < nEnt; e += blockDim.x) {
    smem[e] = gsrc[e];
  }
#endif
}

// ---------------------------------------------------------------------------
// Kernel 3: WMMA nearest-neighbor + collision count.
// One wave32 per 16-query tile; workgroup of 8 waves shares LDS-staged anchor
// chunks. Inner loop per 16-anchor block (unrolled x4 so ds_load latency is
// batched and WMMA co-executes with the previous block's min-tracking VALU):
//   ds_load (packed B-operand + a2) -> broadcast a2 into C
//   -> V_WMMA_F32_16X16X4_F32 (D = a2 - 2 q.a) -> component-wise min/argmin.
// ---------------------------------------------------------------------------
__global__ void __launch_bounds__(256)
ca_nn_collision_kernel(const float* __restrict__ query,
                       const float4* __restrict__ pk,
                       const float* __restrict__ anchor,
                       const float* __restrict__ normals,
                       float* __restrict__ out,
                       int B, int Nq, int Na, int nBlk) {
  __shared__ float4 smem[CHUNK_BLKS * 32];

  const int lane = threadIdx.x & 31;
  const int wave = threadIdx.x >> 5;
  const int b = blockIdx.y;
  const int tile = blockIdx.x * (blockDim.x >> 5) + wave;
  const bool active = (tile * 16) < Nq;
  const int q0 = active ? tile * 16 : 0;   // inactive waves shadow tile 0
                                           // (keep EXEC/barriers uniform)
  const int col  = lane & 15;
  const int half = lane >> 4;

  // --- A matrix: -2 * query coords, 16x4 f32, zero-padded K=3 ---
  int qr = q0 + col;
  if (qr >= Nq) qr = Nq - 1;
  const float* qrow = query + ((size_t)b * Nq + qr) * 3;
  v2f A;
  if (half == 0) { A[0] = -2.0f * qrow[0]; A[1] = -2.0f * qrow[1]; }
  else           { A[0] = -2.0f * qrow[2]; A[1] = 0.0f; }

  float bestV[8];
  int   bestI[8];
#pragma unroll
  for (int v = 0; v < 8; ++v) { bestV[v] = 3.4e38f; bestI[v] = 0; }

  const float4* pkb = pk + (size_t)b * nBlk * 32;

  for (int c0 = 0; c0 < nBlk; c0 += CHUNK_BLKS) {
    int nb = nBlk - c0;
    if (nb > CHUNK_BLKS) nb = CHUNK_BLKS;

    ca_stage_chunk(pkb + (size_t)c0 * 32, smem, nb * 32);
    __syncthreads();

#pragma unroll 4
    for (int blk = 0; blk < nb; ++blk) {
      float4 t = smem[blk * 32 + lane];
      v2f Bm;
      Bm[0] = t.x;
      Bm[1] = t.y;
      v8f C;
#pragma unroll
      for (int v = 0; v < 8; ++v) C[v] = t.z;   // a2 broadcast per column

      v8f D = __builtin_amdgcn_wmma_f32_16x16x4_f32(
          false, A, false, Bm, (short)0, C, false, false);

      int aIdx = (c0 + blk) * 16 + col;
#pragma unroll
      for (int v = 0; v < 8; ++v) {
        float d = D[v];
        if (d < bestV[v]) { bestV[v] = d; bestI[v] = aIdx; }
      }
    }
    __syncthreads();   // chunk fully consumed before overwrite
  }

  // --- per-row argmin across the 16 lanes of each half-wave ---
#pragma unroll
  for (int v = 0; v < 8; ++v) {
    float val = bestV[v];
    int   idx = bestI[v];
#pragma unroll
    for (int m = 8; m >= 1; m >>= 1) {
      float ov = __shfl_xor(val, m, 16);
      int   oi = __shfl_xor(idx, m, 16);
      if (ov < val) { val = ov; idx = oi; }
    }
    bestV[v] = val;
    bestI[v] = idx;
  }

  // --- lanes 0 and 16 own rows M=v and M=v+8: finish collision test ---
  if (active && col == 0) {
    const float* abase = anchor + (size_t)b * Na * 3;
    const float* nbase = normals + (size_t)b * Na * 3;
    float count = 0.0f;
#pragma unroll
    for (int v = 0; v < 8; ++v) {
      int qi = q0 + v + 8 * half;
      if (qi >= Nq) continue;
      const float* qp = query + ((size_t)b * Nq + qi) * 3;
      int ai = bestI[v];
      const float* ap = abase + (size_t)ai * 3;
      const float* np = nbase + (size_t)ai * 3;
      float dx = qp[0] - ap[0];
      float dy = qp[1] - ap[1];
      float dz = qp[2] - ap[2];
      float l2 = sqrtf(dx * dx + dy * dy + dz * dz);
      float dot = (dx * np[0] + dy * np[1] + dz * np[2]) *
                  (l2 <= MAX_DIST_F ? 1.0f : 0.0f);
      if (dot < 0.0f) count += 1.0f;
    }
    atomicAdd(&out[b], count);
  }
}

// ---------------------------------------------------------------------------
extern "C" void kernel_launch(void* const* d_in, const int* in_sizes, int n_in,
                              void* d_out, int out_size, void* d_ws, size_t ws_size,
                              hipStream_t stream) {
  const float* query  = (const float*)d_in[0];  // (B, Nq, 3) f32
  const float* anchor = (const float*)d_in[1];  // (B, Na, 3) f32
  const int*   faces  = (const int*)d_in[2];    // (F, 3) i32
  float* out = (float*)d_out;                   // (B, 1) f32

  const int B  = out_size;
  const int Nq = in_sizes[0] / (3 * B);
  const int Na = in_sizes[1] / (3 * B);
  const int F  = in_sizes[2] / 3;
  const int nBlk = (Na + 15) / 16;

  // workspace layout: [acc: B*Na*3 f32][cnt: Na f32][pad][pk: B*nBlk*32 float4]
  float* acc = (float*)d_ws;
  float* cnt = acc + (size_t)B * Na * 3;
  size_t off = ((size_t)B * Na * 3 + Na) * sizeof(float);
  off = (off + 15) & ~(size_t)15;
  float4* pk = (float4*)((char*)d_ws + off);
  (void)ws_size;

  // 0) zero accumulators, counts, output
  {
    int total = B * Na * 3 + Na + B;
    ca_init_kernel<<<(total + 255) / 256, 256, 0, stream>>>(
        acc, cnt, out, B * Na * 3, Na, B);
  }
  // 1) face normals scatter
  {
    int total = B * F;
    ca_face_normals_kernel<<<(total + 255) / 256, 256, 0, stream>>>(
        anchor, faces, acc, cnt, B, Na, F);
  }
  // 2) normalize vertex normals (in place in acc)
  {
    int total = B * Na;
    ca_normalize_kernel<<<(total + 255) / 256, 256, 0, stream>>>(
        acc, cnt, B, Na);
  }
  // 2.5) pack anchors into WMMA-ready lane records
  {
    int total = B * nBlk * 32;
    ca_pack_kernel<<<(total + 255) / 256, 256, 0, stream>>>(
        anchor, pk, B, Na, nBlk);
  }
  // 3) WMMA nearest neighbor + collision count (LDS-staged anchor chunks)
  {
    int tiles = (Nq + 15) / 16;
    int wavesPerBlock = 256 / 32;
    dim3 grid((tiles + wavesPerBlock - 1) / wavesPerBlock, B);
    ca_nn_collision_kernel<<<grid, 256, 0, stream>>>(
        query, pk, anchor, acc, out, B, Nq, Na, nBlk);
  }
}